// ImageGNN_19404662243653
// MI455X (gfx1250) — compile-verified
//
#include <hip/hip_runtime.h>

typedef __attribute__((ext_vector_type(2))) float v2f;
typedef __attribute__((ext_vector_type(8))) float v8f;

#define N_NODES   100000
#define N_EDGES   1600000
#define FDIM      64
#define N_GRAPHS  256
#define N_CLASSES 10

// ---------------- degree / normalization ----------------

__global__ void deg_init_k(float* __restrict__ deg) {
    int i = blockIdx.x * blockDim.x + threadIdx.x;
    if (i < N_NODES) deg[i] = 1.0f;            // self-loop contributes 1
}

__global__ void deg_accum_k(const int* __restrict__ dst, float* __restrict__ deg) {
    int e = blockIdx.x * blockDim.x + threadIdx.x;
    if (e < N_EDGES) atomicAdd(&deg[dst[e]], 1.0f);
}

__global__ void dinv_k(float* __restrict__ deg) {
    int i = blockIdx.x * blockDim.x + threadIdx.x;
    if (i < N_NODES) deg[i] = rsqrtf(deg[i]);  // deg >= 1 guaranteed by self-loop
}

// ---------------- dense GEMM via V_WMMA_F32_16X16X4_F32 ----------------
// H[n_rows x 64] = (relu_in ? relu(X) : X) @ W[64 x 64]
// Block = 256 threads = 8 waves; each block does 128 rows, each wave a 16x64 strip.
__global__ __launch_bounds__(256) void gcn_gemm_k(const float* __restrict__ X,
                                                  const float* __restrict__ W,
                                                  float* __restrict__ H,
                                                  int n_rows, int relu_in) {
    __shared__ float As[128][65];   // +1 pad: row-stride 65 -> conflict-free column reads
    __shared__ float Ws[64][65];

    const int t    = threadIdx.x;
    const int row0 = blockIdx.x * 128;

    // cooperative, coalesced A load (zero-pad tail rows, optional fused relu)
    for (int i = 0; i < 32; ++i) {
        int idx = t + i * 256;           // 0..8191
        int r = idx >> 6, c = idx & 63;
        int gr = row0 + r;
        float v = (gr < n_rows) ? X[(size_t)gr * FDIM + c] : 0.0f;
        if (relu_in) v = fmaxf(v, 0.0f);
        As[r][c] = v;
    }
    // W is 64x64
    for (int i = 0; i < 16; ++i) {
        int idx = t + i * 256;           // 0..4095
        Ws[idx >> 6][idx & 63] = W[idx];
    }
    __syncthreads();

    const int lane = t & 31;
    const int wave = t >> 5;
    const int m    = lane & 15;          // M (for A) / N (for B,C) index within tile
    const int h2   = (lane >> 4) << 1;   // K offset: 0 for lanes 0-15, 2 for 16-31

    v8f acc0 = {}, acc1 = {}, acc2 = {}, acc3 = {};
    const int ar = wave * 16 + m;        // A row for this lane

    for (int k = 0; k < FDIM; k += 4) {
        v2f a;                            // A 16x4: VGPR v -> K = v + h2
        a.x = As[ar][k + h2];
        a.y = As[ar][k + h2 + 1];
        v2f b0, b1, b2, b3;               // B 4x16: VGPR v -> K = v + h2, N = m
        b0.x = Ws[k + h2][m];        b0.y = Ws[k + h2 + 1][m];
        b1.x = Ws[k + h2][16 + m];   b1.y = Ws[k + h2 + 1][16 + m];
        b2.x = Ws[k + h2][32 + m];   b2.y = Ws[k + h2 + 1][32 + m];
        b3.x = Ws[k + h2][48 + m];   b3.y = Ws[k + h2 + 1][48 + m];
        acc0 = __builtin_amdgcn_wmma_f32_16x16x4_f32(false, a, false, b0, (short)0, acc0, false, false);
        acc1 = __builtin_amdgcn_wmma_f32_16x16x4_f32(false, a, false, b1, (short)0, acc1, false, false);
        acc2 = __builtin_amdgcn_wmma_f32_16x16x4_f32(false, a, false, b2, (short)0, acc2, false, false);
        acc3 = __builtin_amdgcn_wmma_f32_16x16x4_f32(false, a, false, b3, (short)0, acc3, false, false);
    }

    // C/D layout: VGPR r, lanes 0-15 -> M=r, lanes 16-31 -> M=r+8, N = m
    const int rbase = row0 + wave * 16 + ((lane >> 4) << 3);
    for (int r = 0; r < 8; ++r) {
        int gr = rbase + r;
        if (gr < n_rows) {
            float* hp = H + (size_t)gr * FDIM;
            hp[m]      = acc0[r];
            hp[16 + m] = acc1[r];
            hp[32 + m] = acc2[r];
            hp[48 + m] = acc3[r];
        }
    }
}

// ---------------- aggregation ----------------

// agg[n][c] = bias[c] + dinv[n]^2 * h[n][c]   (bias + fused self-loop term)
__global__ void agg_init_k(const float* __restrict__ h, const float* __restrict__ dinv,
                           const float* __restrict__ bias, float* __restrict__ agg) {
    int idx = blockIdx.x * blockDim.x + threadIdx.x;
    if (idx < N_NODES * FDIM) {
        int n = idx >> 6, c = idx & 63;
        float d = dinv[n];
        agg[idx] = bias[c] + d * d * h[idx];
    }
}

// 16 threads per edge, 4 columns each: gather (16B, L2-resident) + 4 f32 atomics
__global__ void edge_scatter_k(const int* __restrict__ src, const int* __restrict__ dst,
                               const float* __restrict__ dinv, const float* __restrict__ h,
                               float* __restrict__ agg) {
    int idx = blockIdx.x * blockDim.x + threadIdx.x;   // < 25.6M
    int e  = idx >> 4;
    int cg = (idx & 15) << 2;
    if (e < N_EDGES) {
        int s = src[e], d = dst[e];
        float w = dinv[s] * dinv[d];
        const float4 hv = *(const float4*)(h + (size_t)s * FDIM + cg);
        float* ap = agg + (size_t)d * FDIM + cg;
        atomicAdd(ap + 0, w * hv.x);
        atomicAdd(ap + 1, w * hv.y);
        atomicAdd(ap + 2, w * hv.z);
        atomicAdd(ap + 3, w * hv.w);
    }
}

// ---------------- pooling + output ----------------

__global__ void pool_zero_k(float* __restrict__ sums, float* __restrict__ cnt) {
    int i = blockIdx.x * blockDim.x + threadIdx.x;
    if (i < N_GRAPHS * FDIM) sums[i] = 0.0f;
    if (i < N_GRAPHS)        cnt[i]  = 0.0f;
}

__global__ void pool_accum_k(const float* __restrict__ agg, const int* __restrict__ batch,
                             float* __restrict__ sums, float* __restrict__ cnt) {
    int idx = blockIdx.x * blockDim.x + threadIdx.x;   // N_NODES*16 threads
    int n  = idx >> 4;
    int cg = (idx & 15) << 2;
    if (n < N_NODES) {
        int g = batch[n];
        const float4 hv = *(const float4*)(agg + (size_t)n * FDIM + cg);
        float* sp = sums + g * FDIM + cg;
        atomicAdd(sp + 0, fmaxf(hv.x, 0.0f));          // relu fused on read
        atomicAdd(sp + 1, fmaxf(hv.y, 0.0f));
        atomicAdd(sp + 2, fmaxf(hv.z, 0.0f));
        atomicAdd(sp + 3, fmaxf(hv.w, 0.0f));
        if (cg == 0) atomicAdd(&cnt[g], 1.0f);
    }
}

__global__ void final_k(const float* __restrict__ sums, const float* __restrict__ cnt,
                        const float* __restrict__ Wout, const float* __restrict__ bout,
                        float* __restrict__ out) {
    int idx = blockIdx.x * blockDim.x + threadIdx.x;
    if (idx < N_GRAPHS * N_CLASSES) {
        int g = idx / N_CLASSES, c = idx % N_CLASSES;
        float inv = 1.0f / fmaxf(cnt[g], 1.0f);
        float acc = bout[c];
        const float* sp = sums + g * FDIM;
        #pragma unroll
        for (int k = 0; k < FDIM; ++k)
            acc = fmaf(sp[k] * inv, Wout[k * N_CLASSES + c], acc);
        out[idx] = acc;
    }
}

// ---------------- launch ----------------

extern "C" void kernel_launch(void* const* d_in, const int* in_sizes, int n_in,
                              void* d_out, int out_size, void* d_ws, size_t ws_size,
                              hipStream_t stream) {
    const float* x    = (const float*)d_in[0];
    const int*   ei   = (const int*)d_in[1];       // edge_index [2, E] flat
    const int*   src  = ei;
    const int*   dst  = ei + N_EDGES;
    const int*   batch= (const int*)d_in[2];
    const float* W1   = (const float*)d_in[3];
    const float* b1   = (const float*)d_in[4];
    const float* W2   = (const float*)d_in[5];
    const float* b2   = (const float*)d_in[6];
    const float* Wout = (const float*)d_in[7];
    const float* bout = (const float*)d_in[8];
    float*       out  = (float*)d_out;

    float* ws   = (float*)d_ws;
    float* dinv = ws;                                   // 100000 (padded to 102400)
    float* bufA = ws + 102400;                          // h  : N_NODES*64
    float* bufB = bufA + (size_t)N_NODES * FDIM;        // agg: N_NODES*64
    float* sums = bufB + (size_t)N_NODES * FDIM;        // 256*64
    float* cnt  = sums + N_GRAPHS * FDIM;               // 256

    const int nodeBlk = (N_NODES + 255) / 256;
    const int edgeBlk = (N_EDGES + 255) / 256;
    const int featBlk = (N_NODES * FDIM + 255) / 256;
    const int scatBlk = (N_EDGES * 16 + 255) / 256;
    const int gemmBlk = (N_NODES + 127) / 128;

    // symmetric normalization D^{-1/2}
    deg_init_k <<<nodeBlk, 256, 0, stream>>>(dinv);
    deg_accum_k<<<edgeBlk, 256, 0, stream>>>(dst, dinv);
    dinv_k     <<<nodeBlk, 256, 0, stream>>>(dinv);

    // layer 1
    gcn_gemm_k     <<<gemmBlk, 256, 0, stream>>>(x, W1, bufA, N_NODES, 0);
    agg_init_k     <<<featBlk, 256, 0, stream>>>(bufA, dinv, b1, bufB);
    edge_scatter_k <<<scatBlk, 256, 0, stream>>>(src, dst, dinv, bufA, bufB);

    // layer 2 (relu of layer-1 fused into GEMM input load)
    gcn_gemm_k     <<<gemmBlk, 256, 0, stream>>>(bufB, W2, bufA, N_NODES, 1);
    agg_init_k     <<<featBlk, 256, 0, stream>>>(bufA, dinv, b2, bufB);
    edge_scatter_k <<<scatBlk, 256, 0, stream>>>(src, dst, dinv, bufA, bufB);

    // global mean pool (relu fused on read) + output layer
    pool_zero_k  <<<(N_GRAPHS * FDIM + 255) / 256, 256, 0, stream>>>(sums, cnt);
    pool_accum_k <<<(N_NODES * 16 + 255) / 256, 256, 0, stream>>>(bufB, batch, sums, cnt);
    final_k      <<<(N_GRAPHS * N_CLASSES + 255) / 256, 256, 0, stream>>>(sums, cnt, Wout, bout, out);
}